// GAT_1941325218011
// MI455X (gfx1250) — compile-verified
//
#include <hip/hip_runtime.h>
#include <math.h>

// GATv2 x3 for MI455X (gfx1250). fp32 end-to-end; GEMMs via V_WMMA_F32_16X16X4_F32.

typedef __attribute__((ext_vector_type(2))) float v2f;
typedef __attribute__((ext_vector_type(8))) float v8f;

#define ENC_NEG_INF 0x007FFFFFu  // enc_ord(-inf)

__device__ __forceinline__ unsigned enc_ord(float f) {
  unsigned u = __float_as_uint(f);
  return (u & 0x80000000u) ? ~u : (u | 0x80000000u);
}
__device__ __forceinline__ float dec_ord(unsigned u) {
  return (u & 0x80000000u) ? __uint_as_float(u ^ 0x80000000u) : __uint_as_float(~u);
}

// ---------------------------------------------------------------------------
// Y[row, n] = sum_k X[row,k] * W[k,n] + b[n]     (K fixed = 128, row-major)
// Block = NWAVES waves. Block owns a 16-row X tile (LDS, loaded once as
// dwordx4); wave w owns output columns [16w, 16w+16).
// A layout (ISA 7.12.2, 32-bit A 16x4): lanes 0-15 -> M, K={0,1}; lanes
// 16-31 -> K={2,3}. B fed as B^T in the same pattern (lane = N, half = K-pair).
// C/D: VGPR r -> rows {r, r+8} by lane half, N = lane&15.
// NCOLS is a template constant so W addressing folds to immediate offsets and
// the K loop fully unrolls into a straight WMMA chain (no EXEC churn).
// ---------------------------------------------------------------------------
template <int NCOLS, int NWAVES>
__global__ __launch_bounds__(NWAVES * 32)
void gemm128_wmma(const float* __restrict__ X, const float* __restrict__ W,
                  const float* __restrict__ bias, float* __restrict__ Y,
                  int nrows) {
  __shared__ float ldsX[16 * 128];
  const int tid  = threadIdx.x;
  const int row0 = blockIdx.x * 16;

  if (row0 + 16 <= nrows) {                       // fast path: full tile
    const float4* s4 = (const float4*)(X + (size_t)row0 * 128);
    float4* l4 = (float4*)ldsX;
    for (int idx = tid; idx < 512; idx += NWAVES * 32) l4[idx] = s4[idx];
  } else {
    for (int idx = tid; idx < 2048; idx += NWAVES * 32) {
      int r = idx >> 7;
      ldsX[idx] = (row0 + r < nrows) ? X[(size_t)(row0 + r) * 128 + (idx & 127)] : 0.0f;
    }
  }
  __syncthreads();

  const int wave = tid >> 5;
  const int lane = tid & 31;
  const int half = lane >> 4;
  const int m    = lane & 15;
  const int col0 = wave * 16;
  if (col0 >= NCOLS) return;                      // wave-uniform

  const int n = col0 + m;
  int   nc  = n;
  float msk = 1.0f;
  if constexpr ((NCOLS & 15) != 0) {              // only tail tile of 40-col GEMM
    if (n >= NCOLS) { nc = NCOLS - 1; msk = 0.0f; }
  }
  const int koff = 2 * half;
  const float* __restrict__ wp = W + (size_t)koff * NCOLS + nc;  // + kk*NCOLS
  const float* __restrict__ ap = ldsX + m * 128 + koff;          // + kk

  v8f acc = {};
  #pragma unroll
  for (int kk = 0; kk < 128; kk += 4) {
    v2f a, b;
    a[0] = ap[kk];
    a[1] = ap[kk + 1];
    float b0 = wp[kk * NCOLS];
    float b1 = wp[(kk + 1) * NCOLS];
    if constexpr ((NCOLS & 15) != 0) { b0 *= msk; b1 *= msk; }
    b[0] = b0; b[1] = b1;
    acc = __builtin_amdgcn_wmma_f32_16x16x4_f32(false, a, false, b,
                                                (short)0, acc, false, false);
  }

  if (n < NCOLS) {
    float bn = bias[n];
    #pragma unroll
    for (int r = 0; r < 8; ++r) {
      int row = row0 + r + 8 * half;
      if (row < nrows) Y[(size_t)row * NCOLS + n] = acc[r] + bn;
    }
  }
}

// ---------------------------------------------------------------------------
// Edge pass A: alpha[e,h] = att[h,:] . leaky_relu(xl[src]+xr[dst]); seg-max
// via ordered-uint atomicMax. One wave32 per edge, lane = channel.
// ---------------------------------------------------------------------------
__global__ void edge_alpha_max(const float* __restrict__ xl, const float* __restrict__ xr,
                               const long long* __restrict__ src,
                               const long long* __restrict__ dst,
                               const float* __restrict__ att,
                               float* __restrict__ alpha, unsigned* __restrict__ m_enc,
                               int nedges, int heads, int ch, int feat) {
  int e = (blockIdx.x * blockDim.x + threadIdx.x) >> 5;
  int lane = threadIdx.x & 31;
  if (e >= nedges) return;
  long s = (long)src[e];
  long d = (long)dst[e];
  const float* pl = xl + s * feat;
  const float* pr = xr + d * feat;
  for (int h = 0; h < heads; ++h) {
    float part = 0.0f;
    for (int c = lane; c < ch; c += 32) {
      float v = pl[h * ch + c] + pr[h * ch + c];
      v = (v > 0.0f) ? v : 0.2f * v;
      part += v * att[h * ch + c];
    }
    for (int off = 16; off > 0; off >>= 1) part += __shfl_xor(part, off, 32);
    if (lane == 0) {
      alpha[(long)e * heads + h] = part;
      atomicMax(&m_enc[d * heads + h], enc_ord(part));
    }
  }
}

// decode ordered-uint max -> float, non-finite -> 0 (matches reference)
__global__ void decode_max(unsigned* __restrict__ m, long count) {
  long i = (long)blockIdx.x * blockDim.x + threadIdx.x;
  if (i < count) {
    float f = dec_ord(m[i]);
    if (isnan(f) || isinf(f)) f = 0.0f;
    ((float*)m)[i] = f;
  }
}

// ---------------------------------------------------------------------------
// Edge pass B: a = exp(alpha - m[dst]); denom[dst] += a;
// accum[dst,:] += xl[src,:] * a   (normalization deferred to node pass)
// ---------------------------------------------------------------------------
__global__ void edge_scatter(const float* __restrict__ xl,
                             const float* __restrict__ alpha, const float* __restrict__ mmax,
                             const long long* __restrict__ src,
                             const long long* __restrict__ dst,
                             float* __restrict__ accum, float* __restrict__ denom,
                             int nedges, int heads, int ch, int feat) {
  int e = (blockIdx.x * blockDim.x + threadIdx.x) >> 5;
  int lane = threadIdx.x & 31;
  if (e >= nedges) return;
  long s = (long)src[e];
  long d = (long)dst[e];
  const float* pl = xl + s * feat;
  float* pa = accum + d * feat;
  for (int h = 0; h < heads; ++h) {
    float a = expf(alpha[(long)e * heads + h] - mmax[d * heads + h]);
    if (lane == 0) atomicAdd(&denom[d * heads + h], a);
    for (int c = lane; c < ch; c += 32)
      atomicAdd(&pa[h * ch + c], pl[h * ch + c] * a);
  }
}

// ---------------------------------------------------------------------------
// Node finalize (layers 1,2): v = accum/(denom+eps)+bias; LayerNorm(128);
// exact GELU; + residual. One wave32 per node, 4 channels/lane (lane+32*head).
// ---------------------------------------------------------------------------
__global__ void node_finalize(const float* __restrict__ accum, const float* __restrict__ denom,
                              const float* __restrict__ bias, const float* __restrict__ g,
                              const float* __restrict__ be, const float* __restrict__ resid,
                              float* __restrict__ xout, int nnodes) {
  int n = (blockIdx.x * blockDim.x + threadIdx.x) >> 5;
  int lane = threadIdx.x & 31;
  if (n >= nnodes) return;
  float v[4];
  #pragma unroll
  for (int i = 0; i < 4; ++i) {
    int c = lane + 32 * i;                       // head = i
    v[i] = accum[(long)n * 128 + c] / (denom[(long)n * 4 + i] + 1e-16f) + bias[c];
  }
  float sum = v[0] + v[1] + v[2] + v[3];
  for (int off = 16; off > 0; off >>= 1) sum += __shfl_xor(sum, off, 32);
  float mu = sum * (1.0f / 128.0f);
  float var = 0.0f;
  #pragma unroll
  for (int i = 0; i < 4; ++i) { float dv = v[i] - mu; var += dv * dv; }
  for (int off = 16; off > 0; off >>= 1) var += __shfl_xor(var, off, 32);
  float rstd = rsqrtf(var * (1.0f / 128.0f) + 1e-5f);
  #pragma unroll
  for (int i = 0; i < 4; ++i) {
    int c = lane + 32 * i;
    float y = (v[i] - mu) * rstd * g[c] + be[c];
    float ge = 0.5f * y * (1.0f + erff(y * 0.70710678118654752f));
    xout[(long)n * 128 + c] = ge + resid[(long)n * 128 + c];
  }
}

// Layer-3 finalize: out = accum/(denom+eps) + bias   (heads=1, mean == identity)
__global__ void node_out3(const float* __restrict__ accum, const float* __restrict__ denom,
                          const float* __restrict__ bias, float* __restrict__ out, int nnodes) {
  long i = (long)blockIdx.x * blockDim.x + threadIdx.x;
  if (i < (long)nnodes * 40) {
    long n = i / 40; int c = (int)(i % 40);
    out[i] = accum[i] / (denom[n] + 1e-16f) + bias[c];
  }
}

__global__ void fill_u32(unsigned* __restrict__ p, unsigned v, long count) {
  long i = (long)blockIdx.x * blockDim.x + threadIdx.x;
  if (i < count) p[i] = v;
}

static inline void launch_fill(unsigned* p, unsigned v, long cnt, hipStream_t s) {
  int blocks = (int)((cnt + 255) / 256);
  fill_u32<<<blocks, 256, 0, s>>>(p, v, cnt);
}

extern "C" void kernel_launch(void* const* d_in, const int* in_sizes, int n_in,
                              void* d_out, int out_size, void* d_ws, size_t ws_size,
                              hipStream_t stream) {
  (void)n_in; (void)out_size; (void)ws_size;
  const float*     x    = (const float*)d_in[0];
  const long long* ei   = (const long long*)d_in[1];
  const float* W0   = (const float*)d_in[2];  const float* b0    = (const float*)d_in[3];
  const float* W1l  = (const float*)d_in[4];  const float* b1l   = (const float*)d_in[5];
  const float* W1r  = (const float*)d_in[6];  const float* b1r   = (const float*)d_in[7];
  const float* att1 = (const float*)d_in[8];  const float* bias1 = (const float*)d_in[9];
  const float* g1   = (const float*)d_in[10]; const float* be1   = (const float*)d_in[11];
  const float* W2l  = (const float*)d_in[12]; const float* b2l   = (const float*)d_in[13];
  const float* W2r  = (const float*)d_in[14]; const float* b2r   = (const float*)d_in[15];
  const float* att2 = (const float*)d_in[16]; const float* bias2 = (const float*)d_in[17];
  const float* g2   = (const float*)d_in[18]; const float* be2   = (const float*)d_in[19];
  const float* W3l  = (const float*)d_in[20]; const float* b3l   = (const float*)d_in[21];
  const float* W3r  = (const float*)d_in[22]; const float* b3r   = (const float*)d_in[23];
  const float* att3 = (const float*)d_in[24]; const float* bias3 = (const float*)d_in[25];

  const int N = in_sizes[0] / 128;
  const int E = in_sizes[1] / 2;
  const long long* src = ei;
  const long long* dst = ei + E;

  float* ws = (float*)d_ws;
  size_t o = 0;
  float* xl    = ws + o; o += (size_t)N * 128;   // source-side transform
  float* xr    = ws + o; o += (size_t)N * 128;   // target-side transform
  float* acc   = ws + o; o += (size_t)N * 128;   // message accumulator
  float* bufA  = ws + o; o += (size_t)N * 128;   // identity, then x2
  float* bufB  = ws + o; o += (size_t)N * 128;   // x1
  float* alpha = ws + o; o += (size_t)E * 4;
  float* mbuf  = ws + o; o += (size_t)N * 4;     // encoded max, then float max
  float* den   = ws + o; o += (size_t)N * 4;

  const int gemmBlocks = (N + 15) / 16;
  const int edgeBlocks = (E + 7) / 8;            // 1 wave32 per edge
  const int nodeBlocks = (N + 7) / 8;

  // ---- identity = x @ W0 + b0 ----
  gemm128_wmma<128, 8><<<gemmBlocks, 256, 0, stream>>>(x, W0, b0, bufA, N);

  // ---- GAT layer 1 ----
  gemm128_wmma<128, 8><<<gemmBlocks, 256, 0, stream>>>(x, W1l, b1l, xl, N);
  gemm128_wmma<128, 8><<<gemmBlocks, 256, 0, stream>>>(x, W1r, b1r, xr, N);
  launch_fill((unsigned*)mbuf, ENC_NEG_INF, (long)N * 4, stream);
  launch_fill((unsigned*)den, 0u, (long)N * 4, stream);
  launch_fill((unsigned*)acc, 0u, (long)N * 128, stream);
  edge_alpha_max<<<edgeBlocks, 256, 0, stream>>>(xl, xr, src, dst, att1, alpha,
                                                 (unsigned*)mbuf, E, 4, 32, 128);
  decode_max<<<(N * 4 + 255) / 256, 256, 0, stream>>>((unsigned*)mbuf, (long)N * 4);
  edge_scatter<<<edgeBlocks, 256, 0, stream>>>(xl, alpha, mbuf, src, dst, acc, den,
                                               E, 4, 32, 128);
  node_finalize<<<nodeBlocks, 256, 0, stream>>>(acc, den, bias1, g1, be1, bufA, bufB, N);

  // ---- GAT layer 2 (input/residual x1=bufB, output x2=bufA) ----
  gemm128_wmma<128, 8><<<gemmBlocks, 256, 0, stream>>>(bufB, W2l, b2l, xl, N);
  gemm128_wmma<128, 8><<<gemmBlocks, 256, 0, stream>>>(bufB, W2r, b2r, xr, N);
  launch_fill((unsigned*)mbuf, ENC_NEG_INF, (long)N * 4, stream);
  launch_fill((unsigned*)den, 0u, (long)N * 4, stream);
  launch_fill((unsigned*)acc, 0u, (long)N * 128, stream);
  edge_alpha_max<<<edgeBlocks, 256, 0, stream>>>(xl, xr, src, dst, att2, alpha,
                                                 (unsigned*)mbuf, E, 4, 32, 128);
  decode_max<<<(N * 4 + 255) / 256, 256, 0, stream>>>((unsigned*)mbuf, (long)N * 4);
  edge_scatter<<<edgeBlocks, 256, 0, stream>>>(xl, alpha, mbuf, src, dst, acc, den,
                                               E, 4, 32, 128);
  node_finalize<<<nodeBlocks, 256, 0, stream>>>(acc, den, bias2, g2, be2, bufB, bufA, N);

  // ---- GAT layer 3 (heads=1, ch=40, no LN/GELU/residual) ----
  gemm128_wmma<40, 3><<<gemmBlocks, 96, 0, stream>>>(bufA, W3l, b3l, xl, N);
  gemm128_wmma<40, 3><<<gemmBlocks, 96, 0, stream>>>(bufA, W3r, b3r, xr, N);
  launch_fill((unsigned*)mbuf, ENC_NEG_INF, (long)N, stream);
  launch_fill((unsigned*)den, 0u, (long)N, stream);
  launch_fill((unsigned*)acc, 0u, (long)N * 40, stream);
  edge_alpha_max<<<edgeBlocks, 256, 0, stream>>>(xl, xr, src, dst, att3, alpha,
                                                 (unsigned*)mbuf, E, 1, 40, 40);
  decode_max<<<(N + 255) / 256, 256, 0, stream>>>((unsigned*)mbuf, (long)N);
  edge_scatter<<<edgeBlocks, 256, 0, stream>>>(xl, alpha, mbuf, src, dst, acc, den,
                                               E, 1, 40, 40);
  node_out3<<<(int)(((long)N * 40 + 255) / 256), 256, 0, stream>>>(acc, den, bias3,
                                                                   (float*)d_out, N);
}